// GraphAttConv_34668976013389
// MI455X (gfx1250) — compile-verified
//
#include <hip/hip_runtime.h>
#include <hip/hip_bf16.h>

typedef __attribute__((ext_vector_type(2))) float v2f;
typedef __attribute__((ext_vector_type(8))) float v8f;

#define F_IN   512
#define HEADS  8
#define DHEAD  32
#define COUT   256           // HEADS * DHEAD
#define LDSROW 516           // 512 + 4 pad floats -> conflict-free LDS rows

// ---------------------------------------------------------------------------
// Kernel: Wt[c][f] = W[h][f][d] with c = h*32+d  (256 x 512, 0.5 MB)
// Makes WMMA B-fragments contiguous 8-byte loads.
// ---------------------------------------------------------------------------
__global__ void gat_transpose_w(const float* __restrict__ W,
                                float* __restrict__ Wt) {
  const int gid = blockIdx.x * blockDim.x + threadIdx.x;
  if (gid >= COUT * F_IN) return;
  const int c = gid >> 9;        // / 512
  const int f = gid & 511;
  Wt[gid] = W[((size_t)(c >> 5) * F_IN + f) * DHEAD + (c & 31)];
}

// ---------------------------------------------------------------------------
// Kernel: h = x @ Wp  via V_WMMA_F32_16X16X4_F32.
// Block = 256 threads (8 waves); wave w owns head w (two 16x16 tiles).
// A tile (16x512) staged in LDS; B from transposed Wt (b64 per fragment).
// ---------------------------------------------------------------------------
__global__ void gat_gemm_h(const float* __restrict__ x,
                           const float* __restrict__ Wt,
                           float* __restrict__ hbuf, int N) {
  __shared__ float As[16 * LDSROW];
  const int mbase = blockIdx.x * 16;

  for (int s = threadIdx.x; s < 16 * (F_IN / 4); s += blockDim.x) {
    const int row = s >> 7;
    const int q   = s & 127;
    float4 v = make_float4(0.f, 0.f, 0.f, 0.f);
    if (mbase + row < N)
      v = ((const float4*)(x + (size_t)(mbase + row) * F_IN))[q];
    *(float4*)&As[row * LDSROW + q * 4] = v;
  }
  __syncthreads();

  const int lane = threadIdx.x & 31;
  const int w    = threadIdx.x >> 5;   // wave id == head id
  const int half = lane >> 4;
  const int l15  = lane & 15;

  const float* arow  = &As[l15 * LDSROW];
  const float* wrow0 = Wt + (size_t)(w * DHEAD + l15) * F_IN; // col tile 0
  const float* wrow1 = wrow0 + (size_t)16 * F_IN;             // col tile 1

  v8f acc0 = {};
  v8f acc1 = {};

  for (int k0 = 0; k0 < F_IN; k0 += 4) {
    const int ka = k0 + half * 2;
    v2f av;
    av.x = arow[ka];
    av.y = arow[ka + 1];
    v2f b0 = *(const v2f*)(wrow0 + ka);   // B[ka][c], B[ka+1][c]
    v2f b1 = *(const v2f*)(wrow1 + ka);

    acc0 = __builtin_amdgcn_wmma_f32_16x16x4_f32(
        false, av, false, b0, (short)0, acc0, false, false);
    acc1 = __builtin_amdgcn_wmma_f32_16x16x4_f32(
        false, av, false, b1, (short)0, acc1, false, false);
  }

  const int rowbase = mbase + half * 8;
  const int colbase = w * DHEAD + l15;
  for (int r = 0; r < 8; ++r) {
    const int row = rowbase + r;
    if (row < N) {
      const size_t off = (size_t)row * COUT;
      hbuf[off + colbase]      = acc0[r];
      hbuf[off + colbase + 16] = acc1[r];
    }
  }
}

// ---------------------------------------------------------------------------
// Kernel: s_src / s_dst per (node, head)
// ---------------------------------------------------------------------------
__global__ void gat_scores(const float* __restrict__ hbuf,
                           const float* __restrict__ a,
                           float* __restrict__ ssrc,
                           float* __restrict__ sdst, int N) {
  const int gid = blockIdx.x * blockDim.x + threadIdx.x;
  if (gid >= N * HEADS) return;
  const int n  = gid >> 3;
  const int hh = gid & 7;
  const float* hp = hbuf + (size_t)n * COUT + hh * DHEAD;
  const float* ap = a + hh * 2 * DHEAD;
  float ss = 0.f, sd = 0.f;
#pragma unroll
  for (int dd = 0; dd < DHEAD; ++dd) {
    const float hv = hp[dd];
    ss += hv * ap[dd];
    sd += hv * ap[DHEAD + dd];
  }
  ssrc[gid] = ss;
  sdst[gid] = sd;
}

// ---------------------------------------------------------------------------
// CSR build: zero counts -> count -> scan -> fill
// ---------------------------------------------------------------------------
__global__ void csr_zero(int* __restrict__ cnt, int N) {
  const int gid = blockIdx.x * blockDim.x + threadIdx.x;
  if (gid < N) cnt[gid] = 0;
}

__global__ void csr_count(const int* __restrict__ e0, int* __restrict__ cnt,
                          int E) {
  const int gid = blockIdx.x * blockDim.x + threadIdx.x;
  if (gid < E) atomicAdd(&cnt[e0[gid]], 1);
}

// single-block Hillis-Steele scan over N elements (exclusive prefix)
__global__ void csr_scan(const int* __restrict__ cnt, int* __restrict__ rowptr,
                         int* __restrict__ cursor, int N) {
  __shared__ int smem[1024];
  __shared__ int running;
  if (threadIdx.x == 0) running = 0;
  __syncthreads();
  for (int base = 0; base < N; base += 1024) {
    const int i = base + (int)threadIdx.x;
    const int v = (i < N) ? cnt[i] : 0;
    const int r0 = running;
    smem[threadIdx.x] = v;
    __syncthreads();
    for (int off = 1; off < 1024; off <<= 1) {
      const int t = (threadIdx.x >= (unsigned)off) ? smem[threadIdx.x - off] : 0;
      __syncthreads();
      smem[threadIdx.x] += t;
      __syncthreads();
    }
    const int incl  = smem[threadIdx.x];
    const int total = smem[1023];
    if (i < N) {
      const int p = r0 + incl - v;
      rowptr[i] = p;
      cursor[i] = p;
    }
    __syncthreads();
    if (threadIdx.x == 0) running = r0 + total;
    __syncthreads();
  }
  if (threadIdx.x == 0) rowptr[N] = running;
}

__global__ void csr_fill(const int* __restrict__ e0, int* __restrict__ cursor,
                         int* __restrict__ eidx, int E) {
  const int gid = blockIdx.x * blockDim.x + threadIdx.x;
  if (gid < E) {
    const int pos = atomicAdd(&cursor[e0[gid]], 1);
    eidx[pos] = gid;
  }
}

// ---------------------------------------------------------------------------
// Kernel: per-destination-node softmax + gather-accumulate. No atomics.
// Block = 256 threads = 256 output channels; 1 block per node.
// Phase A: per-head max, Phase B: per-head exp-sum, Phase C: coalesced
// accumulation of alpha * h[e1] with per-chunk alpha staged in LDS.
// ---------------------------------------------------------------------------
__global__ void gat_node(const int* __restrict__ e1,
                         const int* __restrict__ rowptr,
                         const int* __restrict__ eidx,
                         const float* __restrict__ ssrc,
                         const float* __restrict__ sdst,
                         const float* __restrict__ hbuf,
                         float* __restrict__ out) {
  const int n   = blockIdx.x;
  const int tid = threadIdx.x;
  const int beg = rowptr[n];
  const int deg = rowptr[n + 1] - beg;

  __shared__ float red[256];   // [j=32][hh=8] = red[tid]
  __shared__ float mh[HEADS], dh[HEADS], ssh[HEADS];
  __shared__ int   tsh[32];

  if (tid < HEADS) ssh[tid] = ssrc[n * HEADS + tid];
  __syncthreads();

  const int hh = tid & 7;      // head for reduction phases
  const int j  = tid >> 3;     // edge slot 0..31

  // Phase A: per-head max of leaky-relu logits
  float lm = -3.0e38f;
  for (int k = j; k < deg; k += 32) {
    const int t = e1[eidx[beg + k]];
    float v = ssh[hh] + sdst[t * HEADS + hh];
    v = (v >= 0.f) ? v : 0.2f * v;
    lm = fmaxf(lm, v);
  }
  red[tid] = lm;
  __syncthreads();
  for (int s = 128; s >= 8; s >>= 1) {
    if (tid < s) red[tid] = fmaxf(red[tid], red[tid + s]);
    __syncthreads();
  }
  if (tid < HEADS) mh[tid] = red[tid];
  __syncthreads();

  // Phase B: per-head sum of exp(l - m)
  float se = 0.f;
  for (int k = j; k < deg; k += 32) {
    const int t = e1[eidx[beg + k]];
    float v = ssh[hh] + sdst[t * HEADS + hh];
    v = (v >= 0.f) ? v : 0.2f * v;
    se += __expf(v - mh[hh]);
  }
  red[tid] = se;
  __syncthreads();
  for (int s = 128; s >= 8; s >>= 1) {
    if (tid < s) red[tid] += red[tid + s];
    __syncthreads();
  }
  if (tid < HEADS) dh[tid] = red[tid] + 1e-16f;
  __syncthreads();

  // Phase C: out[n, c] = sum_e alpha[e, head(c)] * h[e1, c]
  const int hc = tid >> 5;     // head of this channel
  float acc = 0.f;
  for (int base = 0; base < deg; base += 32) {
    const int cnt = min(32, deg - base);
    if (j < cnt) {
      const int t = e1[eidx[beg + base + j]];
      if (hh == 0) tsh[j] = t;
      float v = ssh[hh] + sdst[t * HEADS + hh];
      v = (v >= 0.f) ? v : 0.2f * v;
      red[j * 8 + hh] = __expf(v - mh[hh]) / dh[hh];   // alpha
    }
    __syncthreads();
    for (int k = 0; k < cnt; ++k) {
      const int t = tsh[k];
      acc += red[k * 8 + hc] * hbuf[(size_t)t * COUT + tid];
    }
    __syncthreads();
  }
  out[(size_t)n * COUT + tid] = acc;
}

// ---------------------------------------------------------------------------
extern "C" void kernel_launch(void* const* d_in, const int* in_sizes, int n_in,
                              void* d_out, int out_size, void* d_ws,
                              size_t ws_size, hipStream_t stream) {
  const float* x  = (const float*)d_in[0];
  const int*   ei = (const int*)d_in[1];    // (2, E) int32
  const float* W  = (const float*)d_in[2];  // (8, 512, 32)
  const float* a  = (const float*)d_in[3];  // (8, 64)
  float* out = (float*)d_out;

  const int N = in_sizes[0] / F_IN;
  const int E = in_sizes[1] / 2;
  const int* e0 = ei;
  const int* e1 = ei + E;

  // workspace layout
  float* hbuf = (float*)d_ws;                      // N*256 f32
  float* ssrc = hbuf + (size_t)N * COUT;           // N*8
  float* sdst = ssrc + (size_t)N * HEADS;          // N*8
  float* Wt   = sdst + (size_t)N * HEADS;          // 256*512
  int*   cnt    = (int*)(Wt + (size_t)COUT * F_IN);// N
  int*   rowptr = cnt + N;                         // N+1
  int*   cursor = rowptr + (N + 1);                // N
  int*   eidx   = cursor + N;                      // E

  // 1) transpose weights for contiguous WMMA B fragments
  gat_transpose_w<<<(COUT * F_IN + 255) / 256, 256, 0, stream>>>(W, Wt);
  // 2) GEMM h = x @ Wp via f32 WMMA
  gat_gemm_h<<<(N + 15) / 16, 256, 0, stream>>>(x, Wt, hbuf, N);
  // 3) per-node attention scores
  gat_scores<<<(N * HEADS + 255) / 256, 256, 0, stream>>>(hbuf, a, ssrc, sdst, N);
  // 4) CSR by destination e0
  csr_zero<<<(N + 255) / 256, 256, 0, stream>>>(cnt, N);
  csr_count<<<(E + 255) / 256, 256, 0, stream>>>(e0, cnt, E);
  csr_scan<<<1, 1024, 0, stream>>>(cnt, rowptr, cursor, N);
  csr_fill<<<(E + 255) / 256, 256, 0, stream>>>(e0, cursor, eidx, E);
  // 5) fused softmax + message aggregation (atomic-free output)
  gat_node<<<N, COUT, 0, stream>>>(e1, rowptr, eidx, ssrc, sdst, hbuf, out);
}